// HPGNN_81312320848271
// MI455X (gfx1250) — compile-verified
//
#include <hip/hip_runtime.h>
#include <math.h>

#define G_   256
#define M_   256
#define N_   (G_*M_)
#define E_   (N_*16)
#define C_   3
#define K_   5
#define D_   128

typedef float v2f __attribute__((ext_vector_type(2)));
typedef float v8f __attribute__((ext_vector_type(8)));

// ---- workspace layout (float offsets) ----
enum : int {
  OFF_U0    = 0,                      // C*G*M
  OFF_MK    = OFF_U0  + C_*G_*M_,     // C*G*M
  OFF_U1    = OFF_MK  + C_*G_*M_,     // C*G*M
  OFF_A0    = OFF_U1  + C_*G_*M_,     // G*M
  OFF_AK    = OFF_A0  + G_*M_,        // G*M
  OFF_A1    = OFF_AK  + G_*M_,        // G*M
  OFF_UKD   = OFF_A1  + G_*M_,        // C*G
  OFF_AKD   = OFF_UKD + C_*G_,        // G
  OFF_PINFO = OFF_AKD + G_,           // C*256 bins
  OFF_PL2   = OFF_PINFO + C_*256,     // C*256 bins
  OFF_LPATH = OFF_PL2   + C_*256,     // C
  WS_FLOATS = OFF_LPATH + C_
};

__global__ void zero_ws(float* __restrict__ ws) {
  int i = blockIdx.x * 256 + threadIdx.x;
  if (i < WS_FLOATS) ws[i] = 0.0f;
}

// ---- pass 1: scatter row-s / col-t / (s,t) entries of Mdj (per class) and Adj,
//              fused L_info / L_l2 reductions over sij ----
__global__ void edge_pass1(const int* __restrict__ ei, const float* __restrict__ sij,
                           const int* __restrict__ s, const int* __restrict__ t,
                           float* __restrict__ ws) {
  int e = blockIdx.x * 256 + threadIdx.x;
  float linfo[C_] = {0.f, 0.f, 0.f};
  float l2[C_]    = {0.f, 0.f, 0.f};
  if (e < E_) {
    int src = ei[e];
    int dst = ei[E_ + e];
    int g = src >> 8;            // batch[src] = src // M  (M = 256)
    int r = src & 255;
    int c = dst - (g << 8);
    int sg = s[g], tg = t[g];
    bool isr = (r == sg), isc = (c == tg);
    if (isr) atomicAdd(&ws[OFF_A0 + g*M_ + c], 1.0f);
    if (isc) atomicAdd(&ws[OFF_AK + g*M_ + r], 1.0f);
    if (isr && isc) atomicAdd(&ws[OFF_AKD + g], 1.0f);
#pragma unroll
    for (int cc = 0; cc < C_; ++cc) {
      float v = sij[(size_t)cc * E_ + e];
      if (isr) atomicAdd(&ws[OFF_U0 + (cc*G_ + g)*M_ + c], v);
      if (isc) atomicAdd(&ws[OFF_MK + (cc*G_ + g)*M_ + r], v);
      if (isr && isc) atomicAdd(&ws[OFF_UKD + cc*G_ + g], v);
      // infoloss term, r = 0.5
      linfo[cc] = v * logf(2.0f*v + 1e-6f)
                + (1.0f - v) * logf((1.0f - v) / (0.5f + 1e-6f) + 1e-6f);
      l2[cc] = v * v;
    }
  }
  // wave(32) + block reduction, then binned atomics (256 bins)
  __shared__ float sred[8][6];
  int lane = threadIdx.x & 31, wid = threadIdx.x >> 5;
  float vals[6] = {linfo[0], linfo[1], linfo[2], l2[0], l2[1], l2[2]};
#pragma unroll
  for (int j = 0; j < 6; ++j) {
    float v = vals[j];
#pragma unroll
    for (int o = 16; o > 0; o >>= 1) v += __shfl_down(v, o, 32);
    if (lane == 0) sred[wid][j] = v;
  }
  __syncthreads();
  if (wid == 0 && lane < 6) {
    int j = lane;
    float acc = 0.f;
#pragma unroll
    for (int w = 0; w < 8; ++w) acc += sred[w][j];
    int bin = blockIdx.x & 255;
    float* base = (j < 3) ? &ws[OFF_PINFO + j*256 + bin]
                          : &ws[OFF_PL2 + (j-3)*256 + bin];
    atomicAdd(base, acc);
  }
}

// ---- pass 2: one hop:  u1 = u0 @ Mdj  (per class),  a1 = a0 @ Adj ----
__global__ void edge_pass2(const int* __restrict__ ei, const float* __restrict__ sij,
                           float* __restrict__ ws) {
  int e = blockIdx.x * 256 + threadIdx.x;
  if (e >= E_) return;
  int src = ei[e];
  int dst = ei[E_ + e];
  int g = src >> 8;
  int r = src & 255;
  int c = dst - (g << 8);
  float av = ws[OFF_A0 + g*M_ + r];
  if (av != 0.0f) atomicAdd(&ws[OFF_A1 + g*M_ + c], av);
#pragma unroll
  for (int cc = 0; cc < C_; ++cc) {
    float u = ws[OFF_U0 + (cc*G_ + g)*M_ + r];
    if (u != 0.0f) {
      float v = sij[(size_t)cc * E_ + e];
      atomicAdd(&ws[OFF_U1 + (cc*G_ + g)*M_ + c], u * v);
    }
  }
}

__device__ __forceinline__ float blockMax256(float v, float* sm) {
  int n = threadIdx.x;
  sm[n] = v;
  __syncthreads();
#pragma unroll
  for (int st = 128; st > 0; st >>= 1) {
    if (n < st) sm[n] = fmaxf(sm[n], sm[n + st]);
    __syncthreads();
  }
  float r = sm[0];
  __syncthreads();
  return r;
}

// ---- per-group path loss finalize (one block per g, 256 threads = 8 waves) ----
__global__ void path_finalize(const float* __restrict__ pw, float* __restrict__ ws) {
  __shared__ float sm[256];
  int g = blockIdx.x;
  int n = threadIdx.x;
  float w1 = fminf(fmaxf(pw[1], 1e-10f), 1.0f);
  float w2 = fminf(fmaxf(pw[2], 1e-10f), 1.0f);
  float a0v = ws[OFF_A0 + g*M_ + n];
  float akv = ws[OFF_AK + g*M_ + n];
  float a1v = ws[OFF_A1 + g*M_ + n];
  for (int cc = 0; cc < C_; ++cc) {
    float u0v = ws[OFF_U0 + (cc*G_ + g)*M_ + n];
    float mkv = ws[OFF_MK + (cc*G_ + g)*M_ + n];
    float u1v = ws[OFF_U1 + (cc*G_ + g)*M_ + n];
    float p0 = a0v * akv; p0 = (p0 == 0.0f) ? 1e-8f : p0;   // check_zero
    float p1 = a1v * akv; p1 = (p1 == 0.0f) ? 1e-8f : p1;
    float t0 = u0v * mkv / p0;
    float t1 = u1v * mkv / p1;
    float m0 = blockMax256(t0, sm);
    float m1 = blockMax256(t1, sm);
    if (n == 0) {
      float ad = ws[OFF_AKD + g];
      ad = (ad == 0.0f) ? 1e-8f : ad;
      float uk = ws[OFF_UKD + cc*G_ + g] / ad
               + w1 * (m0 + 1e-8f)            // pow(x, 1/1)
               + w2 * sqrtf(m1 + 1e-8f);      // pow(x, 1/2)
      uk *= (1.0f / 3.0f);                    // / MAXLEN
      atomicAdd(&ws[OFF_LPATH + cc], -logf(uk + 1e-8f));
    }
    __syncthreads();
  }
}

__device__ __forceinline__ float blockSum256(float v, float* sm) {
  int n = threadIdx.x;
  sm[n] = v;
  __syncthreads();
#pragma unroll
  for (int st = 128; st > 0; st >>= 1) {
    if (n < st) sm[n] += sm[n + st];
    __syncthreads();
  }
  float r = sm[0];
  __syncthreads();
  return r;
}

__global__ void finalize_scalars(const float* __restrict__ ws, float* __restrict__ out) {
  __shared__ float sm[256];
  int n = threadIdx.x;
  float li = 0.f, l2 = 0.f;
#pragma unroll
  for (int cc = 0; cc < C_; ++cc) {
    li += ws[OFF_PINFO + cc*256 + n];
    l2 += ws[OFF_PL2   + cc*256 + n];
  }
  float LI = blockSum256(li, sm);
  float L2 = blockSum256(l2, sm);
  if (n == 0) {
    float LP = (ws[OFF_LPATH + 0] + ws[OFF_LPATH + 1] + ws[OFF_LPATH + 2])
             / (float)G_ / 3.0f;
    out[C_*G_*D_ + 0] = LI / (float)E_ / 3.0f;          // L_info.mean()
    out[C_*G_*D_ + 1] = LP;                             // L_path.mean()
    out[C_*G_*D_ + 2] = L2 / (2.0f * (float)G_) / 3.0f; // L_l2.mean()
  }
}

// ---- attention: scores = hg @ Hc^T, softmax(K=5), H_emb = att @ Hc
//      fp32 WMMA path: V_WMMA_F32_16X16X4_F32, one wave per 16-row tile.
//      Operands staged in LDS with zero-padding hoisted out of the K-loop,
//      so the loop is pure ds_load_b64 + v_wmma (no exec divergence). ----
__global__ void attn_wmma(const float* __restrict__ hg, const float* __restrict__ H,
                          float* __restrict__ out) {
  int cls = blockIdx.y;
  int g0  = blockIdx.x * 16;
  const float* Hc = H + (size_t)cls * K_ * D_;
  int lane = threadIdx.x;        // 0..31 (wave32)
  int half = lane >> 4;          // K-half select
  int l15  = lane & 15;

  __shared__ __align__(16) float sHc[16 * D_];  // Hc rows 0..4, rows 5..15 = 0
  __shared__ __align__(16) float sHg[16 * D_];  // hg tile, coalesced load
  __shared__ __align__(16) float sc[16 * 16];
  __shared__ __align__(16) float satt[16 * 8];

  // Cooperative, branch-light staging (consecutive lanes -> consecutive addrs)
  for (int idx = lane; idx < 16 * D_; idx += 32) {
    int row = idx >> 7;                       // D_ == 128
    sHc[idx] = (row < K_) ? Hc[idx] : 0.0f;   // idx == row*D_ + col
    sHg[idx] = hg[(size_t)g0 * D_ + idx];
  }
  __syncthreads();

  // scores tile: A = hg[g0:g0+16, :] (16xK), B = Hc^T (Dx16, cols>=5 zero)
  // A frag: lane (half,l15) -> row l15, K = kk + half*2 .. +1
  // B frag: lane (half,l15) -> col l15, K = kk + half*2 .. +1  (B[k][n] = Hc[n][k])
  v8f acc = {0.f,0.f,0.f,0.f,0.f,0.f,0.f,0.f};
#pragma unroll 4
  for (int kk = 0; kk < D_; kk += 4) {
    int kb = kk + half * 2;
    v2f a = *(const v2f*)&sHg[l15 * D_ + kb];
    v2f b = *(const v2f*)&sHc[l15 * D_ + kb];
    acc = __builtin_amdgcn_wmma_f32_16x16x4_f32(false, a, false, b,
                                                (short)0, acc, false, false);
  }
  // D layout: vgpr i holds row (half*8+i), col l15
#pragma unroll
  for (int i = 0; i < 8; ++i) sc[(half * 8 + i) * 16 + l15] = acc[i];
  __syncthreads();

  // softmax over the 5 valid columns; lanes 0..15 each own one row
  if (lane < 16) {
    float v[K_];
    float mx = -1e30f;
#pragma unroll
    for (int j = 0; j < K_; ++j) { v[j] = sc[lane * 16 + j]; mx = fmaxf(mx, v[j]); }
    float ssum = 0.f;
#pragma unroll
    for (int j = 0; j < K_; ++j) { v[j] = expf(v[j] - mx); ssum += v[j]; }
#pragma unroll
    for (int j = 0; j < 8; ++j) satt[lane * 8 + j] = (j < K_) ? v[j] / ssum : 0.0f;
  }
  __syncthreads();

  // H_emb tile: A = att (16x8, K padded 5->8), B = Hc (8xD, rows 5..7 zero)
#pragma unroll
  for (int ct = 0; ct < D_ / 16; ++ct) {
    v8f o = {0.f,0.f,0.f,0.f,0.f,0.f,0.f,0.f};
#pragma unroll
    for (int kk = 0; kk < 8; kk += 4) {
      int kb = kk + half * 2;
      v2f a = *(const v2f*)&satt[l15 * 8 + kb];
      int d = ct * 16 + l15;
      v2f b;
      b.x = sHc[kb * D_ + d];
      b.y = sHc[(kb + 1) * D_ + d];
      o = __builtin_amdgcn_wmma_f32_16x16x4_f32(false, a, false, b,
                                                (short)0, o, false, false);
    }
#pragma unroll
    for (int i = 0; i < 8; ++i)
      out[((size_t)cls * G_ + g0 + half * 8 + i) * D_ + ct * 16 + l15] = o[i];
  }
}

extern "C" void kernel_launch(void* const* d_in, const int* in_sizes, int n_in,
                              void* d_out, int out_size, void* d_ws, size_t ws_size,
                              hipStream_t stream) {
  const float* hg  = (const float*)d_in[0];   // (G, D)
  const float* H   = (const float*)d_in[1];   // (C, K, D)
  const float* pw  = (const float*)d_in[2];   // (MAXLEN,)
  const float* sij = (const float*)d_in[3];   // (C, E)
  const int*   ei  = (const int*)d_in[4];     // (2, E)
  // d_in[5] = batch (derivable: src >> 8), d_in[6] = s, d_in[7] = t
  const int*   s   = (const int*)d_in[6];
  const int*   t   = (const int*)d_in[7];
  float* out = (float*)d_out;
  float* ws  = (float*)d_ws;

  zero_ws<<<(WS_FLOATS + 255) / 256, 256, 0, stream>>>(ws);
  edge_pass1<<<E_ / 256, 256, 0, stream>>>(ei, sij, s, t, ws);
  edge_pass2<<<E_ / 256, 256, 0, stream>>>(ei, sij, ws);
  path_finalize<<<G_, 256, 0, stream>>>(pw, ws);
  finalize_scalars<<<1, 256, 0, stream>>>(ws, out);
  attn_wmma<<<dim3(16, C_), 32, 0, stream>>>(hg, H, out);
}